// VectorQuantizer_71030169141370
// MI455X (gfx1250) — compile-verified
//
#include <hip/hip_runtime.h>
#include <hip/hip_bf16.h>
#include <float.h>

typedef __attribute__((ext_vector_type(2))) float v2f;
typedef __attribute__((ext_vector_type(8))) float v8f;

#define NUM_K 1024
#define DIM   64
#define NTOK  131072
#define DECAYF 0.99f
#define CCOST  0.25f

// ---- workspace layout (float offsets) ----
#define WS_E2   0
#define WS_CNT  1024
#define WS_DW   2048
#define WS_LOSS (2048 + NUM_K*DIM)
#define WS_DMIN (WS_LOSS + 1)

// ---- output layout (float offsets, reference tuple order) ----
#define OFF_LOSS 0ull
#define OFF_Q    1ull
#define OFF_SP   (OFF_Q + (unsigned long long)NTOK*DIM)
#define OFF_PERP (OFF_SP + (unsigned long long)NTOK*NUM_K)
#define OFF_IDX  (OFF_PERP + 1ull)
#define OFF_ACT  (OFF_IDX + (unsigned long long)NTOK)
#define OFF_USE  (OFF_ACT + 1ull)
#define OFF_NCS  (OFF_USE + 1ull)
#define OFF_NW   (OFF_NCS + (unsigned long long)NUM_K)
#define OFF_NE   (OFF_NW + (unsigned long long)NUM_K*DIM)

// K1: zero accumulators + precompute ||e||^2
__global__ __launch_bounds__(256) void vq_init(const float* __restrict__ E,
                                               float* __restrict__ ws) {
  int t = blockIdx.x * 256 + threadIdx.x;
  if (t < NUM_K * DIM) ws[WS_DW + t] = 0.f;
  if (t < NUM_K) {
    ws[WS_CNT + t] = 0.f;
    const float* er = E + (size_t)t * DIM;
    float s = 0.f;
#pragma unroll
    for (int d = 0; d < DIM; ++d) s += er[d] * er[d];
    ws[WS_E2 + t] = s;
  }
  if (t == 0) ws[WS_LOSS] = 0.f;
}

// K2: WMMA distance GEMM + argmin + quantize + EMA scatter accumulation
__global__ __launch_bounds__(256) void vq_dist(const float* __restrict__ X,
                                               const float* __restrict__ E,
                                               float* __restrict__ out,
                                               float* __restrict__ ws) {
  __shared__ float e2s[NUM_K];
  __shared__ float x2s[128];
  __shared__ int   idxs[128];
  __shared__ float dms[128];
  __shared__ float red[8];

  const int tid  = threadIdx.x;
  const int lane = tid & 31;
  const int w    = tid >> 5;
  const int half = lane >> 4;
  const int r    = lane & 15;
  const int m0   = blockIdx.x * 128;   // first token of block
  const int mw   = m0 + w * 16;        // first token of this wave

  for (int i = tid; i < NUM_K; i += 256) e2s[i] = ws[WS_E2 + i];

  // --- A fragments: 16x4 fp32 per k-step; lane r = row, VGPR pair = K pair ---
  v2f a[16];
  float x2p = 0.f;
  const float* xrow = X + (size_t)(mw + r) * DIM;
#pragma unroll
  for (int t = 0; t < 16; ++t) {
    float2 v = *(const float2*)(xrow + 4 * t + 2 * half);
    a[t].x = v.x; a[t].y = v.y;
    x2p += v.x * v.x + v.y * v.y;
  }
  x2p += __shfl_xor(x2p, 16, 32);             // full row norm
  if (!half) x2s[w * 16 + r] = x2p;
  __syncthreads();

  float x2row[8];
#pragma unroll
  for (int j = 0; j < 8; ++j) x2row[j] = x2s[w * 16 + j + 8 * half];

  float minv[8]; int mini[8];
#pragma unroll
  for (int j = 0; j < 8; ++j) { minv[j] = FLT_MAX; mini[j] = 0; }

  float* sp = out + OFF_SP;

  for (int nt = 0; nt < NUM_K / 16; ++nt) {
    const int ncol = nt * 16 + r;
    const float* erow = E + (size_t)ncol * DIM;     // L2-resident codebook
    v2f b[16];
#pragma unroll
    for (int t = 0; t < 16; ++t) {
      float2 v = *(const float2*)(erow + 4 * t + 2 * half);
      b[t].x = v.x; b[t].y = v.y;
    }
    const float e2n = e2s[ncol];
    v8f acc = {0.f, 0.f, 0.f, 0.f, 0.f, 0.f, 0.f, 0.f};
#pragma unroll
    for (int t = 0; t < 16; ++t)
      acc = __builtin_amdgcn_wmma_f32_16x16x4_f32(false, a[t], false, b[t],
                                                  (short)0, acc, false, false);
    // C layout: VGPR j -> row j (lanes 0-15) / row j+8 (lanes 16-31), col = r
#pragma unroll
    for (int j = 0; j < 8; ++j) {
      float d = x2row[j] + e2n - 2.f * acc[j];
      sp[(size_t)(mw + j + 8 * half) * NUM_K + ncol] = d;
      if (d < minv[j]) { minv[j] = d; mini[j] = ncol; }
    }
  }

  // reduce (min, argmin) across the 16 column-lanes of each half
#pragma unroll
  for (int j = 0; j < 8; ++j) {
    float v = minv[j]; int id = mini[j];
#pragma unroll
    for (int off = 8; off > 0; off >>= 1) {
      float ov = __shfl_xor(v, off, 16);
      int   oi = __shfl_xor(id, off, 16);
      if (ov < v || (ov == v && oi < id)) { v = ov; id = oi; }
    }
    if (r == 0) {
      idxs[w * 16 + j + 8 * half] = id;
      dms [w * 16 + j + 8 * half] = v;
    }
  }
  __syncthreads();

  // --- quantize gather, loss partial, dw scatter ---
  float* outq = out + OFF_Q;
  float* cnt  = ws + WS_CNT;
  float* dw   = ws + WS_DW;
  float lsum = 0.f;
  for (int e = tid; e < 128 * DIM; e += 256) {
    int tl = e >> 6, dd = e & 63;
    int id = idxs[tl];
    float x = X[(size_t)(m0 + tl) * DIM + dd];
    float q = E[(size_t)id * DIM + dd];
    outq[(size_t)(m0 + tl) * DIM + dd] = q;   // quantized_ste == quantized fwd
    float df = q - x;
    lsum += df * df;
    atomicAdd(&dw[id * DIM + dd], x);
  }
#pragma unroll
  for (int off = 16; off > 0; off >>= 1) lsum += __shfl_xor(lsum, off, 32);
  if (lane == 0) red[w] = lsum;
  __syncthreads();
  if (tid == 0) {
    float s = 0.f;
#pragma unroll
    for (int i = 0; i < 8; ++i) s += red[i];
    atomicAdd(&ws[WS_LOSS], s);
  }
  for (int tl = tid; tl < 128; tl += 256) {
    int id = idxs[tl];
    atomicAdd(&cnt[id], 1.0f);
    out[OFF_IDX + m0 + tl] = (float)id;
    ws[WS_DMIN + m0 + tl]  = dms[tl];
  }
}

// K3: softmax(-d) per row, in place over the staged distances
__global__ __launch_bounds__(256) void vq_softmax(float* __restrict__ out,
                                                  const float* __restrict__ ws) {
  __shared__ float red[8];
  __shared__ float bsum;
  const int row = blockIdx.x;
  float* p = out + OFF_SP + (size_t)row * NUM_K;
  const float dmin = ws[WS_DMIN + row];    // max of -d == -min d
  float v[4]; float s = 0.f;
#pragma unroll
  for (int i = 0; i < 4; ++i) {
    float d = p[threadIdx.x + 256 * i];
    v[i] = __expf(dmin - d);
    s += v[i];
  }
#pragma unroll
  for (int off = 16; off > 0; off >>= 1) s += __shfl_xor(s, off, 32);
  if ((threadIdx.x & 31) == 0) red[threadIdx.x >> 5] = s;
  __syncthreads();
  if (threadIdx.x == 0) {
    float t = 0.f;
#pragma unroll
    for (int i = 0; i < 8; ++i) t += red[i];
    bsum = t;
  }
  __syncthreads();
  float inv = 1.f / bsum;
#pragma unroll
  for (int i = 0; i < 4; ++i) p[threadIdx.x + 256 * i] = v[i] * inv;
}

__device__ __forceinline__ float blockSum1024(float v, float* red, float* bval) {
#pragma unroll
  for (int off = 16; off > 0; off >>= 1) v += __shfl_xor(v, off, 32);
  __syncthreads();
  if ((threadIdx.x & 31) == 0) red[threadIdx.x >> 5] = v;
  __syncthreads();
  if (threadIdx.x == 0) {
    float t = 0.f;
    for (int i = 0; i < 32; ++i) t += red[i];
    *bval = t;
  }
  __syncthreads();
  return *bval;
}

// K4: EMA update, smoothing, perplexity, scalars
__global__ __launch_bounds__(1024) void vq_final(const float* __restrict__ ema_cs,
                                                 const float* __restrict__ ema_w,
                                                 float* __restrict__ out,
                                                 const float* __restrict__ ws) {
  __shared__ float red[32];
  __shared__ float bval;
  const int k = threadIdx.x;

  float cnt = ws[WS_CNT + k];
  float ncs = DECAYF * ema_cs[k] + (1.f - DECAYF) * cnt;

  float n = blockSum1024(ncs, red, &bval);

  float avg = cnt / (float)NTOK;
  float plsum = blockSum1024(avg * __logf(avg + 1e-10f), red, &bval);
  float actsum = blockSum1024(cnt > 0.f ? 1.f : 0.f, red, &bval);

  float smoothed = (ncs + 1e-5f) / (n + (float)NUM_K * 1e-5f) * n;
  out[OFF_NCS + k] = ncs;
  const float* wrow  = ema_w + (size_t)k * DIM;
  const float* dwrow = ws + WS_DW + k * DIM;
  float invs = 1.f / smoothed;
#pragma unroll
  for (int d = 0; d < DIM; ++d) {
    float nw = DECAYF * wrow[d] + (1.f - DECAYF) * dwrow[d];
    out[OFF_NW + (size_t)k * DIM + d] = nw;
    out[OFF_NE + (size_t)k * DIM + d] = nw * invs;
  }
  if (k == 0) {
    out[OFF_LOSS] = CCOST * ws[WS_LOSS] / (float)(NTOK * DIM);
    out[OFF_PERP] = __expf(-plsum);
    out[OFF_ACT]  = actsum;
    out[OFF_USE]  = actsum / (float)NUM_K * 100.f;
  }
}

extern "C" void kernel_launch(void* const* d_in, const int* in_sizes, int n_in,
                              void* d_out, int out_size, void* d_ws, size_t ws_size,
                              hipStream_t stream) {
  const float* X   = (const float*)d_in[0];  // inputs [32,4096,64]
  const float* E   = (const float*)d_in[1];  // embedding [1024,64]
  const float* ecs = (const float*)d_in[2];  // ema_cluster_size [1024]
  const float* ew  = (const float*)d_in[3];  // ema_w [1024,64]
  float* out = (float*)d_out;
  float* ws  = (float*)d_ws;

  vq_init<<<(NUM_K * DIM + 255) / 256, 256, 0, stream>>>(E, ws);
  vq_dist<<<NTOK / 128, 256, 0, stream>>>(X, E, out, ws);
  vq_softmax<<<NTOK, 256, 0, stream>>>(out, ws);
  vq_final<<<1, 1024, 0, stream>>>(ecs, ew, out, ws);
}